// TopKOperator_7370163880549
// MI455X (gfx1250) — compile-verified
//
#include <hip/hip_runtime.h>
#include <math.h>

typedef __attribute__((ext_vector_type(2))) float v2f;
typedef __attribute__((ext_vector_type(8))) float v8f;

#define NB   16
#define LEN  4096
#define EDIM 1024
#define OUTL 512

// ---------------------------------------------------------------------------
// Phase 1: per batch — bitonic sort of scores in LDS, 3 halving rounds on
// scores, tracking (orig_index, weight) provenance per surviving slot.
// Writes final scores to d_out tail and 128 (idx,w) pairs per 16-slot tile.
// ---------------------------------------------------------------------------
__global__ __launch_bounds__(1024) void topk_phase1(
    const float* __restrict__ scores, float* __restrict__ outScores,
    char* prov_base, unsigned long long prov_stride,
    char* scr_base, unsigned long long scr_stride)
{
  __shared__ float s[LEN];
  __shared__ int   slotid[LEN];
  const int tid = threadIdx.x;
  const int b   = blockIdx.x;

  char*  scr = scr_base + (unsigned long long)b * scr_stride;
  int*   iA  = (int*)  (scr + 0);      // 4096 ints
  float* wA  = (float*)(scr + 16384);  // 4096 floats
  int*   iB  = (int*)  (scr + 32768);
  float* wB  = (float*)(scr + 49152);

  for (int i = tid; i < LEN; i += 1024) { s[i] = scores[b * LEN + i]; slotid[i] = i; }
  __syncthreads();

  int curLen = LEN;
  for (int r = 1; r <= 3; ++r) {
    // ---- bitonic sort, descending by (score desc, slot asc), first curLen ----
    for (int k = 2; k <= curLen; k <<= 1) {
      for (int j = k >> 1; j > 0; j >>= 1) {
        for (int i = tid; i < curLen; i += 1024) {
          int ixj = i ^ j;
          if (ixj > i) {
            float si = s[i], sj = s[ixj];
            int ii = slotid[i], ij = slotid[ixj];
            bool before  = (si > sj) || (si == sj && ii < ij); // strict desc order
            bool dirDesc = ((i & k) == 0);
            if (dirDesc != before) { s[i] = sj; s[ixj] = si; slotid[i] = ij; slotid[ixj] = ii; }
          }
        }
        __syncthreads();
      }
    }
    // ---- merge: pair rank i with rank (curLen-1-i)  (FLIP_RIGHT fold) ----
    int half  = curLen >> 1;
    int csize = 1 << (r - 1);
    const int*   srcI = (r == 2) ? iB : iA;
    const float* srcW = (r == 2) ? wB : wA;
    int*   dstI = (r == 2) ? iA : iB;
    float* dstW = (r == 2) ? wA : wB;
    for (int i = tid; i < half; i += 1024) {
      float sc0 = s[i],        sc1 = s[curLen - 1 - i];
      int   a   = slotid[i],   bb  = slotid[curLen - 1 - i];
      // conv = softmax(2^sc) over the pair
      float p0 = exp2f(sc0), p1 = exp2f(sc1);
      float mx = fmaxf(p0, p1);
      float e0 = expf(p0 - mx), e1 = expf(p1 - mx);
      float inv = 1.0f / (e0 + e1);
      float w0 = e0 * inv, w1 = e1 * inv;
      int base = i * 2 * csize;
      if (r == 1) {
        dstI[base]     = a;  dstW[base]     = w0;
        dstI[base + 1] = bb; dstW[base + 1] = w1;
      } else {
        for (int j = 0; j < csize; ++j) {
          dstI[base + j]         = srcI[a  * csize + j];
          dstW[base + j]         = srcW[a  * csize + j] * w0;
          dstI[base + csize + j] = srcI[bb * csize + j];
          dstW[base + csize + j] = srcW[bb * csize + j] * w1;
        }
      }
      s[i]      = sc0 * w0 + sc1 * w1;
      slotid[i] = i;
    }
    __threadfence_block();
    __syncthreads();
    curLen = half;
  }

  // final scores (pair order of round 3 == reference output order)
  if (tid < OUTL) outScores[b * OUTL + tid] = s[tid];
  // provenance: block (b,st) = 128 ints then 128 floats (slots st*16..st*16+15)
  for (int e = tid; e < 4096; e += 1024) {
    int st = e >> 7, local = e & 127;
    char* pb = prov_base + (unsigned long long)(b * 32 + st) * prov_stride;
    ((int*)pb)[local]         = iB[e];
    ((float*)pb)[128 + local] = wB[e];
  }
}

// ---------------------------------------------------------------------------
// Phase 2: one workgroup per (batch, 16-slot tile). Gather the 128
// contributing rows via global_load_async_to_lds_b128 into per-wave LDS and
// accumulate D(16x16) = A(16x128 block-diag weights) x B(128x16 gathered)
// with V_WMMA_F32_16X16X4_F32 (32 K-chunks of 4).
// ---------------------------------------------------------------------------
#define P2_WAVES 4
__global__ __launch_bounds__(128) void topk_phase2(
    const float* __restrict__ embs, float* __restrict__ outEmb,
    const char* prov_base, unsigned long long prov_stride)
{
  // single static LDS object -> base offset 0, so integer LDS addresses below are valid
  __shared__ alignas(16) char smem[1024 + P2_WAVES * 8192];
  const int tid = threadIdx.x;
  const int b   = blockIdx.x >> 5;
  const int st  = blockIdx.x & 31;

  int*   gidxp = (int*)smem;            // [128] gathered row indices
  float* gwp   = (float*)(smem + 512);  // [128] weights
  const char* pb = prov_base + (unsigned long long)(b * 32 + st) * prov_stride;
  if (tid < 128) {
    gidxp[tid] = ((const int*)pb)[tid];
    gwp[tid]   = ((const float*)pb)[128 + tid];
  }
  __syncthreads();

  const int wave  = tid >> 5;
  const int lane  = tid & 31;
  const int n     = lane & 15;   // A-row / B-col / D-col index for this lane
  const int halfw = lane >> 4;   // lane half selects K pair (2h, 2h+1) of each chunk per ISA layout

  // A row n has its 8 weights at K = 8n..8n+7 -> nonzero only in chunks 2n, 2n+1
  float wa0 = gwp[8 * n + 2 * halfw];
  float wa1 = gwp[8 * n + 2 * halfw + 1];
  float wb0 = gwp[8 * n + 4 + 2 * halfw];
  float wb1 = gwp[8 * n + 4 + 2 * halfw + 1];

  const int rowL = lane >> 2;    // 0..7  (rows covered per async group)
  const int seg  = lane & 3;     // 0..3  (16B segments of a 64B row-slice)
  float* Bsw = (float*)(smem + 1024 + (wave << 13)); // this wave's 128x16 stage
  const unsigned bsOff = 1024u + (unsigned)(wave << 13);
  const size_t embBatch = (size_t)b * LEN * EDIM;
  const size_t outBase0 = ((size_t)b * OUTL + st * 16) * EDIM;

  for (int ct = wave; ct < 64; ct += P2_WAVES) {   // 64 column tiles of 16
    int n0 = ct << 4;
    // async gather: 128 rows x 64B into this wave's LDS stage
    for (int g = 0; g < 16; ++g) {
      int rrow = (g << 3) + rowL;
      int gi = gidxp[rrow];
      unsigned long long srcAddr =
          (unsigned long long)(embs + embBatch + ((size_t)gi << 10) + n0 + (seg << 2));
      unsigned lds = bsOff + (unsigned)((rrow << 6) + (seg << 4));
      asm volatile("global_load_async_to_lds_b128 %0, %1, off"
                   :: "v"(lds), "v"(srcAddr) : "memory");
    }
    asm volatile("s_wait_asynccnt 0x0" ::: "memory");

    v8f acc = {0.f, 0.f, 0.f, 0.f, 0.f, 0.f, 0.f, 0.f};
    for (int kk = 0; kk < 32; ++kk) {
      int kg0 = (kk << 2) + (halfw << 1);  // K for VGPR0 of this lane-half
      v2f a2; a2.x = 0.f; a2.y = 0.f;
      if (kk == (n << 1))          { a2.x = wa0; a2.y = wa1; }
      else if (kk == (n << 1) + 1) { a2.x = wb0; a2.y = wb1; }
      v2f b2;
      b2.x = Bsw[(kg0 << 4) + n];        // B[k][n], row stride 16 floats
      b2.y = Bsw[((kg0 + 1) << 4) + n];
      acc = __builtin_amdgcn_wmma_f32_16x16x4_f32(false, a2, false, b2,
                                                  (short)0, acc, false, false);
    }
    // D layout: VGPR r -> M=r (lanes 0-15) / M=r+8 (lanes 16-31), lane = N
    for (int r2 = 0; r2 < 8; ++r2) {
      int mm = r2 + (halfw << 3);
      outEmb[outBase0 + (size_t)mm * EDIM + n0 + n] = acc[r2];
    }
  }
}

// ---------------------------------------------------------------------------
extern "C" void kernel_launch(void* const* d_in, const int* in_sizes, int n_in,
                              void* d_out, int out_size, void* d_ws, size_t ws_size,
                              hipStream_t stream) {
  const float* embs   = (const float*)d_in[0];
  const float* scores = (const float*)d_in[1];
  float* outEmb    = (float*)d_out;
  float* outScores = outEmb + (size_t)NB * OUTL * EDIM;

  // provenance: 16*32 blocks of 1KB (128 idx + 128 w); scratch: 64KB per batch
  char* prov_base; unsigned long long prov_stride;
  char* scr_base;  unsigned long long scr_stride;
  if (ws_size >= (size_t)(NB * 32 * 1024 + NB * 65536)) {
    prov_base = (char*)d_ws;          prov_stride = 1024;
    scr_base  = (char*)d_ws + 524288; scr_stride  = 65536;
  } else {
    // stash inside d_out's embedding region (fully overwritten by phase 2;
    // each phase-2 workgroup reads its own 1KB block before writing its tile)
    prov_base = (char*)d_out;          prov_stride = 65536;
    scr_base  = (char*)d_out + 131072; scr_stride  = 2097152;
  }

  topk_phase1<<<16, 1024, 0, stream>>>(scores, outScores,
                                       prov_base, prov_stride, scr_base, scr_stride);
  topk_phase2<<<NB * 32, 128, 0, stream>>>(embs, outEmb, prov_base, prov_stride);
}